// GCNBackbone_45853070852694
// MI455X (gfx1250) — compile-verified
//
#include <hip/hip_runtime.h>
#include <hip/hip_bf16.h>

typedef __attribute__((ext_vector_type(2))) float v2f;
typedef __attribute__((ext_vector_type(8))) float v8f;

#define D 128
#define LDSW 132  // padded LDS row stride (floats) to avoid bank aliasing

// ---------------------------------------------------------------------------
// deg/dinv setup
// ---------------------------------------------------------------------------
__global__ void deg_init_kernel(float* __restrict__ deg, int n) {
    int i = blockIdx.x * blockDim.x + threadIdx.x;
    if (i < n) deg[i] = 1.0f;  // self-loop contributes 1 to every target
}

__global__ void deg_accum_kernel(const int* __restrict__ ei, float* __restrict__ deg, int E) {
    int e = blockIdx.x * blockDim.x + threadIdx.x;
    if (e < E) atomicAdd(&deg[ei[(size_t)E + e]], 1.0f);  // dst row of edge_index
}

__global__ void dinv_kernel(float* __restrict__ dinv, int n) {
    int i = blockIdx.x * blockDim.x + threadIdx.x;
    if (i < n) dinv[i] = rsqrtf(dinv[i]);  // deg >= 1 always
}

// ---------------------------------------------------------------------------
// fp32 GEMM  C[n x 128] = (RELU_IN ? relu(A) : A)[n x 128] @ W[128 x 128]
// 8 waves/block, wave = one 16-row slab, V_WMMA_F32_16X16X4_F32.
// RELU_IN is a template param so each layer gets a clean specialization
// (no cndmask blend of both variants).
// ---------------------------------------------------------------------------
template <bool RELU_IN>
__global__ __launch_bounds__(256)
void gemm128_kernel(const float* __restrict__ A, const float* __restrict__ W,
                    float* __restrict__ C, int n) {
    __shared__ float wlds[D * LDSW];

    // Stage W (64 KB) into LDS with padded stride: 128 rows x 32 float4
    for (int i = threadIdx.x; i < D * (D / 4); i += 256) {
        int r = i >> 5;
        int c4 = (i & 31) * 4;
        float4 w = *(const float4*)(W + (size_t)r * D + c4);
        *(float4*)(&wlds[r * LDSW + c4]) = w;
    }
    __syncthreads();

    int wave = threadIdx.x >> 5;
    int lane = threadIdx.x & 31;
    int rowTile = blockIdx.x * 8 + wave;           // 16-row tile index (wave-uniform)
    if (rowTile * 16 >= n) return;                 // uniform exit: EXEC stays all-ones
    int row0 = rowTile * 16;

    // A fragment layout (16x4 fp32): lanes 0-15 hold K=koff,koff+1 for M=lane;
    // lanes 16-31 hold K=koff+2,koff+3 for M=lane-16.
    int m    = lane & 15;
    int koff = (lane >> 4) * 2;
    const float* arow = A + (size_t)(row0 + m) * D + koff;

    v2f afrag[32];
#pragma unroll
    for (int k = 0; k < 32; ++k) {
        float ax = arow[k * 4 + 0];
        float ay = arow[k * 4 + 1];
        if (RELU_IN) { ax = fmaxf(ax, 0.0f); ay = fmaxf(ay, 0.0f); }
        afrag[k].x = ax;
        afrag[k].y = ay;
    }

    for (int nt = 0; nt < 8; ++nt) {               // 8 column tiles of 16
        int col = nt * 16 + (lane & 15);
        v8f acc = {0.f, 0.f, 0.f, 0.f, 0.f, 0.f, 0.f, 0.f};
#pragma unroll
        for (int k = 0; k < 32; ++k) {
            // B fragment (4x16 fp32): lane group selects rows koff / koff+2
            const float* wp = &wlds[(k * 4 + koff) * LDSW + col];
            v2f b;
            b.x = wp[0];
            b.y = wp[LDSW];
            acc = __builtin_amdgcn_wmma_f32_16x16x4_f32(
                /*neg_a=*/false, afrag[k], /*neg_b=*/false, b,
                /*c_mod=*/(short)0, acc, /*reuse_a=*/false, /*reuse_b=*/false);
        }
        // C/D layout: VGPR r -> row row0+r (lanes 0-15) / row0+8+r (lanes 16-31)
        int srow = row0 + (lane >> 4) * 8;
        float* cp = C + (size_t)srow * D + nt * 16 + (lane & 15);
#pragma unroll
        for (int r = 0; r < 8; ++r) cp[(size_t)r * D] = acc[r];
    }
}

// ---------------------------------------------------------------------------
// agg[i] = b + dinv[i]^2 * t[i]   (bias + fused self-loop message)
// one thread = 4 channels of one node
// ---------------------------------------------------------------------------
__global__ void init_agg_kernel(const float* __restrict__ t, const float* __restrict__ bias,
                                const float* __restrict__ dinv, float* __restrict__ agg, int n) {
    int idx = blockIdx.x * blockDim.x + threadIdx.x;     // over n*32
    if (idx >= n * 32) return;
    int node = idx >> 5;
    int c4 = (idx & 31) * 4;
    float di = dinv[node];
    float s = di * di;
    float4 v = *(const float4*)(t + (size_t)node * D + c4);
    float4 b = *(const float4*)(bias + c4);
    float4 o;
    o.x = b.x + v.x * s;
    o.y = b.y + v.y * s;
    o.z = b.z + v.z * s;
    o.w = b.w + v.w * s;
    *(float4*)(agg + (size_t)node * D + c4) = o;
}

// ---------------------------------------------------------------------------
// edge scatter: one wave per edge; 32 lanes x float4 = full 128-float row.
// agg[dst] += dinv[src]*dinv[dst] * t[src]  via global_atomic_add_f32 (L2-resident)
// ---------------------------------------------------------------------------
__global__ __launch_bounds__(256)
void edge_scatter_kernel(const float* __restrict__ t, float* __restrict__ agg,
                         const int* __restrict__ ei, const float* __restrict__ dinv, int E) {
    int gid = blockIdx.x * blockDim.x + threadIdx.x;
    int e = gid >> 5;
    int lane = gid & 31;
    if (e >= E) return;
    int src = ei[e];
    int dst = ei[(size_t)E + e];
    float nrm = dinv[src] * dinv[dst];
    float4 v = *(const float4*)(t + (size_t)src * D + lane * 4);
    float* dp = agg + (size_t)dst * D + lane * 4;
    atomicAdd(dp + 0, v.x * nrm);
    atomicAdd(dp + 1, v.y * nrm);
    atomicAdd(dp + 2, v.z * nrm);
    atomicAdd(dp + 3, v.w * nrm);
}

// ---------------------------------------------------------------------------
// final ReLU in place
// ---------------------------------------------------------------------------
__global__ void relu_kernel(float* __restrict__ p, int n4) {
    int i = blockIdx.x * blockDim.x + threadIdx.x;
    if (i >= n4) return;
    float4 v = *(const float4*)(p + (size_t)i * 4);
    v.x = fmaxf(v.x, 0.f);
    v.y = fmaxf(v.y, 0.f);
    v.z = fmaxf(v.z, 0.f);
    v.w = fmaxf(v.w, 0.f);
    *(float4*)(p + (size_t)i * 4) = v;
}

// ---------------------------------------------------------------------------
extern "C" void kernel_launch(void* const* d_in, const int* in_sizes, int n_in,
                              void* d_out, int out_size, void* d_ws, size_t ws_size,
                              hipStream_t stream) {
    const float* x  = (const float*)d_in[0];
    const float* W1 = (const float*)d_in[1];
    const float* b1 = (const float*)d_in[2];
    const float* W2 = (const float*)d_in[3];
    const float* b2 = (const float*)d_in[4];
    const int*   ei = (const int*)d_in[5];

    const int N = in_sizes[0] / D;
    const int E = in_sizes[5] / 2;
    float* out = (float*)d_out;

    float* t0   = (float*)d_ws;                    // N*128 : GEMM output (messages)
    float* t1   = t0 + (size_t)N * D;              // N*128 : layer-1 aggregate
    float* dinv = t1 + (size_t)N * D;              // N     : deg -> rsqrt(deg)

    const int TPB = 256;
    dim3 blk(TPB);
    int rowTiles = (N + 15) / 16;
    dim3 gemmGrid((rowTiles + 7) / 8);
    dim3 edgeGrid(((size_t)E * 32 + TPB - 1) / TPB);
    dim3 nodeGrid(((size_t)N * 32 + TPB - 1) / TPB);

    // degree / symmetric normalization
    deg_init_kernel<<<(N + TPB - 1) / TPB, blk, 0, stream>>>(dinv, N);
    deg_accum_kernel<<<(E + TPB - 1) / TPB, blk, 0, stream>>>(ei, dinv, E);
    dinv_kernel<<<(N + TPB - 1) / TPB, blk, 0, stream>>>(dinv, N);

    // ---- layer 1 ----
    gemm128_kernel<false><<<gemmGrid, blk, 0, stream>>>(x, W1, t0, N);
    init_agg_kernel<<<nodeGrid, blk, 0, stream>>>(t0, b1, dinv, t1, N);
    edge_scatter_kernel<<<edgeGrid, blk, 0, stream>>>(t0, t1, ei, dinv, E);

    // ---- layer 2 (ReLU of t1 fused into GEMM A-loads) ----
    gemm128_kernel<true><<<gemmGrid, blk, 0, stream>>>(t1, W2, t0, N);
    init_agg_kernel<<<nodeGrid, blk, 0, stream>>>(t0, b2, dinv, out, N);
    edge_scatter_kernel<<<edgeGrid, blk, 0, stream>>>(t0, out, ei, dinv, E);

    relu_kernel<<<((size_t)N * D / 4 + TPB - 1) / TPB, blk, 0, stream>>>(out, N * D / 4);
}